// QueryBasedAttention_88776974008863
// MI455X (gfx1250) — compile-verified
//
#include <hip/hip_runtime.h>

#define NROWS 500000
#define DIM   256
#define HID   256
#define NTILE (NROWS / 16)   // 31250, exact
#define NCHUNK (DIM / 4)     // 64 k-chunks per tile
#define NPART 1024
#define NEG_INF (-__builtin_inff())

// ws layout (float offsets)
#define WS_C    0              // c[256]  (key-side projected query, pre-scaled by 1/16)
#define WS_V    256            // v[256]  (value-side projection of W_out)
#define WS_MS   512            // [0]=global max M, [1]=global sum S
#define WS_PART 1024           // NPART float2 (m,s) partials
#define WS_S    4096           // s[NROWS] (masked scores)
#define WS_U    (4096 + NROWS) // u[NROWS]

typedef __attribute__((ext_vector_type(2))) float v2f;
typedef __attribute__((ext_vector_type(8))) float v8f;

// ---------------- K1: tiny precompute of c and v -----------------
__global__ void k_precompute(const float* __restrict__ Wq, const float* __restrict__ query,
                             const float* __restrict__ Wk, const float* __restrict__ Wv,
                             const float* __restrict__ Wo, float* __restrict__ ws) {
  __shared__ float Q[HID];
  const int t = threadIdx.x;            // 256 threads
  float acc = 0.f;
  for (int d = 0; d < DIM; ++d) acc += Wq[t * DIM + d] * query[d];
  Q[t] = acc;
  __syncthreads();
  float c = 0.f, v = 0.f;
  for (int h = 0; h < HID; ++h) {
    c += Wk[h * DIM + t] * Q[h];
    v += Wv[h * DIM + t] * Wo[h];
  }
  ws[WS_C + t] = c * 0.0625f;           // fold 1/sqrt(256)
  ws[WS_V + t] = v;
}

// ---------------- K2: WMMA streaming pass over inputs -----------------
// Per wave: one 16-row tile. D = sum_k A(16x4) * B(4x16); B col0 = c, col1 = v.
// All 64 B fragments (WMMA register layout) are staged in LDS once per block so
// the hot loop is branch-free: global_load_b64 (A) + ds_load_b64 (B) + wmma.
__global__ void k_main(const float* __restrict__ X, const int* __restrict__ mask,
                       const float* __restrict__ cvec, const float* __restrict__ vvec,
                       float* __restrict__ s_out, float* __restrict__ u_out) {
  // bfrag[kc][lane*2 + j]: B-fragment for k-chunk kc in WMMA per-lane layout.
  // Lane l (n = l&15, kh = l>>4) holds B[kh*2+0][n], B[kh*2+1][n];
  // column 0 = c chunk, column 1 = v chunk, columns 2..15 = 0.
  __shared__ float bfrag[NCHUNK][64];   // 16 KB

  const int tid  = threadIdx.x;
  const int lane = tid & 31;
  const int wave = tid >> 5;

  for (int idx = tid; idx < NCHUNK * 32; idx += blockDim.x) {
    const int kc = idx >> 5;
    const int l  = idx & 31;
    const int n  = l & 15;
    const int k  = kc * 4 + (l >> 4) * 2;
    float bx = 0.f, by = 0.f;
    if (n == 0)      { bx = cvec[k]; by = cvec[k + 1]; }
    else if (n == 1) { bx = vvec[k]; by = vvec[k + 1]; }
    bfrag[kc][l * 2]     = bx;
    bfrag[kc][l * 2 + 1] = by;
  }
  __syncthreads();                      // before any wave can exit

  const int tile = blockIdx.x * (blockDim.x >> 5) + wave;
  if (tile >= NTILE) return;            // wave-uniform; EXEC all-1s for WMMA
  const int row0 = tile * 16;
  const int n  = lane & 15;             // A: row-in-tile, B/C/D: column
  const int kh = lane >> 4;             // k-half selector (ISA 16x4 f32 A layout)
  const float* aptr = X + (size_t)(row0 + n) * DIM + kh * 2;

  v8f acc = (v8f)0.0f;
  #pragma unroll 8
  for (int kc = 0; kc < NCHUNK; ++kc) {
    v2f a = __builtin_nontemporal_load((const v2f*)aptr);  // streaming 512 MB: NT
    aptr += 4;
    v2f b;
    b.x = bfrag[kc][lane * 2];
    b.y = bfrag[kc][lane * 2 + 1];
    acc = __builtin_amdgcn_wmma_f32_16x16x4_f32(
        /*neg_a=*/false, a, /*neg_b=*/false, b,
        /*c_mod=*/(short)0, acc, /*reuse_a=*/false, /*reuse_b=*/false);
  }

  // D layout: lanes 0-15 hold rows M=vgpr (N=lane); lanes 16-31 rows M=vgpr+8.
  if (n < 2) {
    const int rbase = row0 + 8 * kh;
    float vals[8];
    #pragma unroll
    for (int j = 0; j < 8; ++j) vals[j] = acc[j];
    if (n == 0) {                        // scores: apply mask -> -inf
      const int4* mp = (const int4*)(mask + rbase);
      int4 m0 = mp[0], m1 = mp[1];
      if (m0.x == 0) vals[0] = NEG_INF;
      if (m0.y == 0) vals[1] = NEG_INF;
      if (m0.z == 0) vals[2] = NEG_INF;
      if (m0.w == 0) vals[3] = NEG_INF;
      if (m1.x == 0) vals[4] = NEG_INF;
      if (m1.y == 0) vals[5] = NEG_INF;
      if (m1.z == 0) vals[6] = NEG_INF;
      if (m1.w == 0) vals[7] = NEG_INF;
      float4* sp = (float4*)(s_out + rbase);
      sp[0] = make_float4(vals[0], vals[1], vals[2], vals[3]);
      sp[1] = make_float4(vals[4], vals[5], vals[6], vals[7]);
    } else {                             // u = x . v
      float4* up = (float4*)(u_out + rbase);
      up[0] = make_float4(vals[0], vals[1], vals[2], vals[3]);
      up[1] = make_float4(vals[4], vals[5], vals[6], vals[7]);
    }
  }
}

// ---------------- softmax (max, sumexp) deterministic reduction ----------------
__device__ __forceinline__ void smx_combine(float& m1, float& s1, float m2, float s2) {
  float M  = fmaxf(m1, m2);
  float e1 = (m1 == NEG_INF) ? 0.0f : expf(m1 - M);
  float e2 = (m2 == NEG_INF) ? 0.0f : expf(m2 - M);
  s1 = s1 * e1 + s2 * e2;
  m1 = M;
}

__global__ void k_partials(const float* __restrict__ s_in, float2* __restrict__ parts) {
  __shared__ float sm[256], ss[256];
  const int t = threadIdx.x;
  float m = NEG_INF, s = 0.f;
  for (int i = blockIdx.x * blockDim.x + t; i < NROWS; i += gridDim.x * blockDim.x)
    smx_combine(m, s, s_in[i], 1.0f);
  sm[t] = m; ss[t] = s;
  __syncthreads();
  for (int w = 128; w > 0; w >>= 1) {
    if (t < w) {
      float mm = sm[t], sv = ss[t];
      smx_combine(mm, sv, sm[t + w], ss[t + w]);
      sm[t] = mm; ss[t] = sv;
    }
    __syncthreads();
  }
  if (t == 0) parts[blockIdx.x] = make_float2(sm[0], ss[0]);
}

__global__ void k_combine(const float2* __restrict__ parts, float* __restrict__ ms) {
  __shared__ float sm[256], ss[256];
  const int t = threadIdx.x;
  float m = NEG_INF, s = 0.f;
  for (int i = t; i < NPART; i += 256) {
    float2 p = parts[i];
    smx_combine(m, s, p.x, p.y);
  }
  sm[t] = m; ss[t] = s;
  __syncthreads();
  for (int w = 128; w > 0; w >>= 1) {
    if (t < w) {
      float mm = sm[t], sv = ss[t];
      smx_combine(mm, sv, sm[t + w], ss[t + w]);
      sm[t] = mm; ss[t] = sv;
    }
    __syncthreads();
  }
  if (t == 0) { ms[0] = sm[0]; ms[1] = ss[0]; }
}

// ---------------- K5: final elementwise pass ----------------
__global__ void k_output(const float* __restrict__ s_in, const float* __restrict__ u_in,
                         const float* __restrict__ ms, float* __restrict__ out) {
  const int i = blockIdx.x * blockDim.x + threadIdx.x;
  if (i >= NROWS) return;
  const float M = ms[0], S = ms[1];
  const float sv = s_in[i];
  const float w = (sv == NEG_INF) ? 0.0f : expf(sv - M) / S;
  out[i] = 10.0f * tanhf(w * u_in[i]);
}

extern "C" void kernel_launch(void* const* d_in, const int* in_sizes, int n_in,
                              void* d_out, int out_size, void* d_ws, size_t ws_size,
                              hipStream_t stream) {
  const float* inputs = (const float*)d_in[0];
  const float* query  = (const float*)d_in[1];
  const int*   mask   = (const int*)  d_in[2];
  const float* Wq     = (const float*)d_in[3];
  const float* Wk     = (const float*)d_in[4];
  const float* Wv     = (const float*)d_in[5];
  const float* Wo     = (const float*)d_in[6];
  float* ws  = (float*)d_ws;
  float* out = (float*)d_out;

  float*  cvec  = ws + WS_C;
  float*  vvec  = ws + WS_V;
  float*  ms    = ws + WS_MS;
  float2* parts = (float2*)(ws + WS_PART);
  float*  sbuf  = ws + WS_S;
  float*  ubuf  = ws + WS_U;

  k_precompute<<<1, 256, 0, stream>>>(Wq, query, Wk, Wv, Wo, ws);

  const int wavesPerBlock = 8;                       // 256 threads = 8 wave32
  const int blocks2 = (NTILE + wavesPerBlock - 1) / wavesPerBlock;
  k_main<<<blocks2, 32 * wavesPerBlock, 0, stream>>>(inputs, mask, cvec, vvec, sbuf, ubuf);

  k_partials<<<NPART, 256, 0, stream>>>(sbuf, parts);
  k_combine<<<1, 256, 0, stream>>>(parts, ms);
  k_output<<<(NROWS + 255) / 256, 256, 0, stream>>>(sbuf, ubuf, ms, out);
}